// GNNStack_51711406244137
// MI455X (gfx1250) — compile-verified
//
#include <hip/hip_runtime.h>

#define N_NODES   100000
#define N_EDGES   1000000
#define DIM       64
#define OUTD      32
#define NGRAPH    512

typedef __attribute__((ext_vector_type(2))) float v2f;
typedef __attribute__((ext_vector_type(8))) float v8f;
typedef unsigned int u32x4 __attribute__((ext_vector_type(4)));
typedef int          i32x4 __attribute__((ext_vector_type(4)));
typedef int          i32x8 __attribute__((ext_vector_type(8)));

// ---------------------------------------------------------------- utilities
__global__ __launch_bounds__(256) void zero_f32(float* __restrict__ p, int n) {
    int i = blockIdx.x * blockDim.x + threadIdx.x;
    if (i < n) p[i] = 0.0f;
}

// ------------------------------------------------- edge gather + scatter-add
// agg[dst] += x[src]  over 1M edges; 16 threads per edge, float4 per thread.
__global__ __launch_bounds__(256) void scatter_agg(const float* __restrict__ x,
                                                   const int* __restrict__ ei,
                                                   float* __restrict__ agg) {
    long long tid = (long long)blockIdx.x * blockDim.x + threadIdx.x;
    if (tid >= (long long)N_EDGES * 16) return;
    int e  = (int)(tid >> 4);
    int f4 = ((int)tid & 15) * 4;
    int s = ei[e];
    int d = ei[N_EDGES + e];
    const float4 v = *(const float4*)(x + (long long)s * DIM + f4);
    float* o = agg + (long long)d * DIM + f4;
    atomicAdd(o + 0, v.x);
    atomicAdd(o + 1, v.y);
    atomicAdd(o + 2, v.z);
    atomicAdd(o + 3, v.w);
}

// -------------------------------------------------------------- GIN MLP
// Per 16-node tile (one wave32): h1 = relu((x+agg)@W1 + b1); h2 = h1@W2 + b2.
// Full fp32 via V_WMMA_F32_16X16X4_F32.
#define WST 80   // LDS stride for weights: bank = (16k + col) & 63 -> conflict free
#define HST 68   // LDS stride for h1 stage: bank = (4m + col) & 63 -> conflict free

#if defined(__gfx1250__) && __has_builtin(__builtin_amdgcn_tensor_load_to_lds)
#define USE_TDM 1
// TDM: async DMA of a 64x64 fp32 tile from global into LDS, inserting 16 pad
// DWORDs after every 64 DWORDs -> row stride 80 floats (matches WST).
__device__ __forceinline__ void tdm_load_w64x64_pad16(float* lds_dst,
                                                      const float* gsrc) {
    unsigned           lds_off = (unsigned)(unsigned long long)lds_dst; // LDS byte offset
    unsigned long long ga      = (unsigned long long)gsrc;              // 57-bit global addr

    u32x4 g0;
    g0[0] = 1u;                                         // count=1, no gather
    g0[1] = lds_off;                                    // lds_addr
    g0[2] = (unsigned)(ga & 0xffffffffull);             // global_addr[31:0]
    g0[3] = (unsigned)((ga >> 32) & 0x1ffffffull)       // global_addr[56:32]
          | (2u << 30);                                 // type = 2 (image)

    i32x8 g1;
    g1[0] = (2 << 16)                                   // data_size = 4 bytes
          | (1 << 20)                                   // pad_enable
          | (5 << 22)                                   // pad_interval: 2^(5+1)=64 dwords
          | (15 << 25);                                 // pad_amount: 16 dwords
    g1[1] = (64 << 16);                                 // tensor_dim0 = 64
    g1[2] = (64 << 16);                                 // tensor_dim1 = 64
    g1[3] = (64 << 16);                                 // tile_dim0 = 64
    g1[4] = 64;                                         // tile_dim1 = 64, tile_dim2 = 0
    g1[5] = 64;                                         // tensor_dim0_stride = 64
    g1[6] = 0;
    g1[7] = 0;

    i32x4 gz = {0, 0, 0, 0};
#if defined(__clang_major__) && (__clang_major__ >= 23)
    i32x8 gz8 = {0, 0, 0, 0, 0, 0, 0, 0};
    __builtin_amdgcn_tensor_load_to_lds(g0, g1, gz, gz, gz8, 0);
#else
    __builtin_amdgcn_tensor_load_to_lds(g0, g1, gz, gz, 0);
#endif
}
#else
#define USE_TDM 0
#endif

__global__ __launch_bounds__(128) void gin_mlp(const float* __restrict__ x,
                                               const float* __restrict__ agg,
                                               const float* __restrict__ W1,
                                               const float* __restrict__ b1,
                                               const float* __restrict__ W2,
                                               const float* __restrict__ b2,
                                               float* __restrict__ out_act,
                                               float* __restrict__ out_pre) {
    __shared__ float sW1[64 * WST];
    __shared__ float sW2[64 * WST];
    __shared__ float sH[4][16 * HST];

#if USE_TDM
    // One wave issues both weight DMAs; TENSORcnt wait + barrier publishes LDS.
    if (threadIdx.x < 32) {
        tdm_load_w64x64_pad16(sW1, W1);
        tdm_load_w64x64_pad16(sW2, W2);
        __builtin_amdgcn_s_wait_tensorcnt(0);
    }
#else
    for (int i = threadIdx.x; i < 64 * 64; i += 128) {
        int k = i >> 6, n = i & 63;
        sW1[k * WST + n] = W1[i];
        sW2[k * WST + n] = W2[i];
    }
#endif
    __syncthreads();

    const int wave = threadIdx.x >> 5;
    const int lane = threadIdx.x & 31;
    const int row  = lane & 15;     // M row in tile (A) or N col (B/C)
    const int hi   = lane >> 4;     // selects K pair for A/B, +8 rows for C/D

    const int node0 = blockIdx.x * 64 + wave * 16;
    int arow = node0 + row;
    if (arow > N_NODES - 1) arow = N_NODES - 1;   // clamp: garbage confined to unstored rows

    const float* xr = x   + (long long)arow * DIM;
    const float* gr = agg + (long long)arow * DIM;

    // ---- GEMM1: (x+agg)[16x64] @ W1[64x64]
    v8f acc[4] = {};
    for (int k0 = 0; k0 < 64; k0 += 4) {
        const int ka = k0 + hi * 2;
        v2f a;
        a.x = xr[ka]     + gr[ka];
        a.y = xr[ka + 1] + gr[ka + 1];
#pragma unroll
        for (int nt = 0; nt < 4; ++nt) {
            const int ncol = nt * 16 + row;
            v2f b;
            b.x = sW1[ka * WST + ncol];
            b.y = sW1[(ka + 1) * WST + ncol];
            acc[nt] = __builtin_amdgcn_wmma_f32_16x16x4_f32(
                false, a, false, b, (short)0, acc[nt], false, false);
        }
    }

    // bias + relu, re-layout C-tile -> row-major stage in LDS
    float* stage = sH[wave];
#pragma unroll
    for (int nt = 0; nt < 4; ++nt) {
        const int ncol = nt * 16 + row;
        const float bias = b1[ncol];
#pragma unroll
        for (int v = 0; v < 8; ++v) {
            const int m = v + hi * 8;
            stage[m * HST + ncol] = fmaxf(acc[nt][v] + bias, 0.0f);
        }
    }
    __syncthreads();

    // ---- GEMM2: h1[16x64] @ W2[64x64]
    v8f acc2[4] = {};
    for (int k0 = 0; k0 < 64; k0 += 4) {
        const int ka = k0 + hi * 2;
        v2f a;
        a.x = stage[row * HST + ka];
        a.y = stage[row * HST + ka + 1];
#pragma unroll
        for (int nt = 0; nt < 4; ++nt) {
            const int ncol = nt * 16 + row;
            v2f b;
            b.x = sW2[ka * WST + ncol];
            b.y = sW2[(ka + 1) * WST + ncol];
            acc2[nt] = __builtin_amdgcn_wmma_f32_16x16x4_f32(
                false, a, false, b, (short)0, acc2[nt], false, false);
        }
    }

#pragma unroll
    for (int nt = 0; nt < 4; ++nt) {
        const int ncol = nt * 16 + row;
        const float bias = b2[ncol];
#pragma unroll
        for (int v = 0; v < 8; ++v) {
            const int m = v + hi * 8;
            const int nrow = node0 + m;
            if (nrow < N_NODES) {
                const float h = acc2[nt][v] + bias;
                if (out_pre) out_pre[(long long)nrow * DIM + ncol] = h;
                out_act[(long long)nrow * DIM + ncol] = fmaxf(h, 0.0f);
            }
        }
    }
}

// -------------------------------------------------------------- mean pool
__global__ __launch_bounds__(256) void pool_scatter(const float* __restrict__ xact,
                                                    const int* __restrict__ batch,
                                                    float* __restrict__ sums,
                                                    float* __restrict__ cnts) {
    int tid = blockIdx.x * blockDim.x + threadIdx.x;
    if (tid >= N_NODES * 16) return;
    int node = tid >> 4;
    int f4   = (tid & 15) * 4;
    int g = batch[node];
    const float4 v = *(const float4*)(xact + (long long)node * DIM + f4);
    float* o = sums + (long long)g * DIM + f4;
    atomicAdd(o + 0, v.x);
    atomicAdd(o + 1, v.y);
    atomicAdd(o + 2, v.z);
    atomicAdd(o + 3, v.w);
    if (f4 == 0) atomicAdd(cnts + g, 1.0f);
}

// -------------------------------------------------------------- post-MLP
__global__ __launch_bounds__(256) void post_mlp1(const float* __restrict__ sums,
                                                 const float* __restrict__ cnts,
                                                 const float* __restrict__ Wp1,
                                                 const float* __restrict__ bp1,
                                                 float* __restrict__ t) {
    int tid = blockIdx.x * blockDim.x + threadIdx.x;
    if (tid >= NGRAPH * DIM) return;
    int g = tid >> 6;
    int o = tid & 63;
    float inv = 1.0f / fmaxf(cnts[g], 1.0f);
    float a = bp1[o];
    const float* row = sums + g * DIM;
    for (int k = 0; k < DIM; ++k) a += (row[k] * inv) * Wp1[k * DIM + o];
    t[tid] = a;
}

// one wave32 per graph; 32 classes == 32 lanes -> shuffle reductions
__global__ __launch_bounds__(256) void post_mlp2_lsm(const float* __restrict__ t,
                                                     const float* __restrict__ Wp2,
                                                     const float* __restrict__ bp2,
                                                     float* __restrict__ out) {
    int gt = blockIdx.x * blockDim.x + threadIdx.x;
    int g = gt >> 5;
    int lane = threadIdx.x & 31;
    if (g >= NGRAPH) return;
    float a = bp2[lane];
    const float* row = t + g * DIM;
    for (int k = 0; k < DIM; ++k) a += row[k] * Wp2[k * OUTD + lane];
    float m = a;
    for (int off = 16; off > 0; off >>= 1) m = fmaxf(m, __shfl_xor(m, off, 32));
    float e = expf(a - m);
    float s = e;
    for (int off = 16; off > 0; off >>= 1) s += __shfl_xor(s, off, 32);
    out[g * OUTD + lane] = a - m - logf(s);
}

// ---------------------------------------------------------------- driver
extern "C" void kernel_launch(void* const* d_in, const int* in_sizes, int n_in,
                              void* d_out, int out_size, void* d_ws, size_t ws_size,
                              hipStream_t stream) {
    const float* x   = (const float*)d_in[0];
    const int*   ei  = (const int*)d_in[1];     // [2, N_EDGES]: src row then dst row
    const int*   bat = (const int*)d_in[2];
    const float* W1a = (const float*)d_in[3];
    const float* b1a = (const float*)d_in[4];
    const float* W2a = (const float*)d_in[5];
    const float* b2a = (const float*)d_in[6];
    const float* W1b = (const float*)d_in[7];
    const float* b1b = (const float*)d_in[8];
    const float* W2b = (const float*)d_in[9];
    const float* b2b = (const float*)d_in[10];
    const float* Wp1 = (const float*)d_in[11];
    const float* bp1 = (const float*)d_in[12];
    const float* Wp2 = (const float*)d_in[13];
    const float* bp2 = (const float*)d_in[14];

    float* emb = (float*)d_out;                              // [N_NODES*64]
    float* lsm = (float*)d_out + (long long)N_NODES * DIM;   // [512*32]

    const long long NF = (long long)N_NODES * DIM;
    float* ws     = (float*)d_ws;
    float* aggbuf = ws;                  // agg1 -> agg2 -> x2 (safe in-place reuse)
    float* x1     = ws + NF;
    float* sums   = ws + 2 * NF;         // [512*64]
    float* cnts   = sums + NGRAPH * DIM; // [512]
    float* tbuf   = cnts + NGRAPH;       // [512*64]

    const int nf = (int)NF;
    const int zb = (nf + 255) / 256;
    const int eb = (N_EDGES * 16 + 255) / 256;
    const int gb = (N_NODES + 63) / 64;

    // ---- layer 0
    zero_f32<<<zb, 256, 0, stream>>>(aggbuf, nf);
    scatter_agg<<<eb, 256, 0, stream>>>(x, ei, aggbuf);
    gin_mlp<<<gb, 128, 0, stream>>>(x, aggbuf, W1a, b1a, W2a, b2a, x1, (float*)nullptr);

    // ---- layer 1 (emb = pre-relu to d_out, relu output in-place over aggbuf)
    zero_f32<<<zb, 256, 0, stream>>>(aggbuf, nf);
    scatter_agg<<<eb, 256, 0, stream>>>(x1, ei, aggbuf);
    gin_mlp<<<gb, 128, 0, stream>>>(x1, aggbuf, W1b, b1b, W2b, b2b, aggbuf, emb);

    // ---- global mean pool
    const int ps = NGRAPH * DIM + NGRAPH;
    zero_f32<<<(ps + 255) / 256, 256, 0, stream>>>(sums, ps);
    pool_scatter<<<(N_NODES * 16 + 255) / 256, 256, 0, stream>>>(aggbuf, bat, sums, cnts);

    // ---- post-MLP + log_softmax
    post_mlp1<<<(NGRAPH * DIM + 255) / 256, 256, 0, stream>>>(sums, cnts, Wp1, bp1, tbuf);
    post_mlp2_lsm<<<(NGRAPH * 32 + 255) / 256, 256, 0, stream>>>(tbuf, Wp2, bp2, lsm);
}